// PointVectorizer_31044023616316
// MI455X (gfx1250) — compile-verified
//
#include <hip/hip_runtime.h>

#define NCONN 21
#define NPB   256                      // batches per block (8 wave32 waves)
#define FPB   (NCONN * 3)              // 63 floats per batch
#define TILE_DW (NPB * FPB)            // 16128 dwords = 64512 bytes
#define PLANE   (NPB * NCONN)          // 5376 floats per u-component plane

typedef unsigned int       u32;
typedef unsigned long long u64;
typedef __attribute__((ext_vector_type(4))) u32 u32x4;
typedef __attribute__((ext_vector_type(4))) int i32x4;
typedef __attribute__((ext_vector_type(8))) int i32x8;

__device__ __forceinline__ void store_nt_off(float* base, u32 byteOff, float v) {
    // SGPR-64 base + VGPR-32 byte offset -> GLOBAL GVS addressing, 32-bit addr math
    __builtin_nontemporal_store(v, (float*)((char*)base + byteOff));
}

__global__ __launch_bounds__(NPB)
void pv_angles_kernel(const float* __restrict__ lm, float* __restrict__ out, int B) {
    // Phase A/B: raw 256-batch tile (64512 B).  Phase C/D: ux|uy|uz planes (same 64512 B).
    __shared__ float smem[TILE_DW];
    float* const ux = smem;
    float* const uy = smem + PLANE;
    float* const uz = smem + 2 * PLANE;

    const int t           = threadIdx.x;
    const u64 tileBaseF   = (u64)blockIdx.x * (u64)TILE_DW;
    const u64 totalFloats = (u64)B * (u64)FPB;
    const bool fullTile   = (tileBaseF + (u64)TILE_DW) <= totalFloats;

    bool needFallback = true;

#if defined(__has_builtin) && __has_builtin(__builtin_amdgcn_tensor_load_to_lds)
    // -------- Phase A: TDM DMA stages the contiguous 64512-byte slab into LDS ----------
    if (fullTile) {
        needFallback = false;
        if (t < 32) {  // one wave issues the DMA (wave-uniform branch)
            const u64 ga     = (u64)lm + tileBaseF * 4ull;
            const u32 ldsOff = (u32)(size_t)&smem[0];

            // D# group 0: count=1, lds_addr, global_addr[56:0], type=2
            u32x4 g0;
            g0.x = 1u;
            g0.y = ldsOff;
            g0.z = (u32)ga;
            g0.w = ((u32)(ga >> 32) & 0x01FFFFFFu) | (2u << 30);

            // D# group 1: data_size=4B; 1-D tensor/tile of TILE_DW elements
            i32x8 g1 = {
                (int)(2u << 16),
                (int)(((u32)TILE_DW & 0xFFFFu) << 16),
                (int)((((u32)TILE_DW >> 16) & 0xFFFFu) | (1u << 16)),
                (int)(((u32)TILE_DW & 0xFFFFu) << 16),
                (int)1,
                (int)(u32)TILE_DW,
                0,
                0
            };
            i32x4 g2 = { 1, 1, 0, 0 };
            i32x4 g3 = { 0, (int)(1u << 16), 0, 0 };

#if __has_include(<hip/amd_detail/amd_gfx1250_TDM.h>)
            i32x8 gz = { 0, 0, 0, 0, 0, 0, 0, 0 };
            __builtin_amdgcn_tensor_load_to_lds(g0, g1, g2, g3, gz, 0);
#else
            __builtin_amdgcn_tensor_load_to_lds(g0, g1, g2, g3, 0);
#endif
            __builtin_amdgcn_s_wait_tensorcnt(0);
        }
    }
#endif

    if (needFallback) {  // partial last tile (or no TDM builtin): guarded coalesced loads
        const float* src = lm + tileBaseF;
        u64 n = totalFloats - tileBaseF;
        if (n > (u64)TILE_DW) n = (u64)TILE_DW;
        for (u32 i = (u32)t; i < (u32)n; i += NPB) smem[i] = src[i];
    }

    __syncthreads();

    // -------- Phase B: each thread builds its normalized vectors in registers ----------
    float uxr[NCONN], uyr[NCONN], uzr[NCONN];
    {
        const float* p = &smem[t * FPB];           // stride-63 -> LDS conflict-free
        const float x0 = p[0], y0 = p[1], z0 = p[2];
#pragma unroll
        for (int j = 0; j < NCONN; ++j) {
            const float x = p[3 * j + 0] - x0;
            const float y = p[3 * j + 1] - y0;
            const float z = p[3 * j + 2] - z0;
            const float r = 1.0f / __builtin_sqrtf(x * x + y * y + z * z); // j==0 -> inf (u0 never read)
            uxr[j] = x * r; uyr[j] = y * r; uzr[j] = z * r;
        }
    }
    __syncthreads();   // all raw-tile reads complete before planes overwrite the buffer

    // -------- Phase C: write u into three planes (bank-conflict-free columns) ----------
#pragma unroll
    for (int j = 0; j < NCONN; ++j) {
        ux[j * NPB + t] = uxr[j];
        uy[j * NPB + t] = uyr[j];
        uz[j * NPB + t] = uzr[j];
    }
    __syncthreads();

    // -------- Phase D: triangular angle computation + streaming stores -----------------
    const u64 batch = (u64)blockIdx.x * (u64)NPB + (u64)t;
    if (batch >= (u64)B) return;                   // after the last barrier: safe

    const u32 Bu      = (u32)B;
    const u32 rowStep = Bu * 4u;                   // +1 in j along a row  (bytes)
    const u32 colStep = (u32)NCONN * Bu * 4u;      // +1 in j along a column (bytes)
    const u32 base32  = (u32)batch * 4u;           // total output < 2^31 bytes -> u32 offsets

    // Row 0 and column 0 are exactly 0 (v0 == 0 -> 0/0 -> NaN -> 0 in the reference).
#pragma unroll
    for (int j = 0; j < NCONN; ++j)
        store_nt_off(out, (u32)j * rowStep + base32, 0.0f);
#pragma unroll
    for (int i = 1; i < NCONN; ++i)
        store_nt_off(out, (u32)(i * NCONN) * rowStep + base32, 0.0f);

#pragma unroll 1
    for (int i = 1; i < NCONN; ++i) {
        const int ib = i * NPB + t;
        const float xi = ux[ib], yi = uy[ib], zi = uz[ib];

        u32 o1 = (u32)(i * NCONN + i) * rowStep + base32;  // (i,j) walks row i
        u32 o2 = o1;                                        // (j,i) walks column i
        {   // diagonal (i,i): cos ~= 1 (tiny angle or NaN->0, matching reference)
            const float c = xi * xi + yi * yi + zi * zi;
            float a = acosf(c);
            a = (a == a) ? a : 0.0f;
            store_nt_off(out, o1, a);
            o1 += rowStep;
            o2 += colStep;
        }
#pragma unroll 1
        for (int j = i + 1; j < NCONN; ++j) {
            const int jb = j * NPB + t;
            const float c = xi * ux[jb] + yi * uy[jb] + zi * uz[jb];
            float a = acosf(c);                    // c>1 or NaN -> NaN
            a = (a == a) ? a : 0.0f;               // NaN -> 0 (reference semantics)
            store_nt_off(out, o1, a);              // (i,j)
            store_nt_off(out, o2, a);              // (j,i) — symmetric twin
            o1 += rowStep;
            o2 += colStep;
        }
    }
}

extern "C" void kernel_launch(void* const* d_in, const int* in_sizes, int n_in,
                              void* d_out, int out_size, void* d_ws, size_t ws_size,
                              hipStream_t stream) {
    const float* lm = (const float*)d_in[0];
    float* out = (float*)d_out;
    const int B = in_sizes[0] / FPB;               // 262144 for the reference setup
    const int blocks = (B + NPB - 1) / NPB;        // 1024 blocks
    pv_angles_kernel<<<blocks, NPB, 0, stream>>>(lm, out, B);
}